// ChildSumTreeLSTMWithAttention_78554951844439
// MI455X (gfx1250) — compile-verified
//
#include <hip/hip_runtime.h>
#include <hip/hip_bf16.h>

#define HDIM 128
#define EPSV 1e-9f

typedef __attribute__((ext_vector_type(2))) float v2f;
typedef __attribute__((ext_vector_type(8))) float v8f;

// ---------------------------------------------------------------------------
// Y[m, n] = sum_k X[m,k] * W[n,k]  (+ bias[n])  (+ Y if accumulate)
// One wave computes a 16x16 output tile via 32 x V_WMMA_F32_16X16X4_F32.
// Block: 256 threads = 8 waves, covers 128 rows of X; grid.y = H/16 = 8 N-tiles.
// Weight tile (16 outputs x 128 K) staged in padded LDS (conflict-free b64 reads).
//
// WMMA f32 16x16x4 operand layout (ISA 7.12.2, 32-bit A 16x4):
//   A: lanes 0-15 hold row m=lane, K={k,k+1}; lanes 16-31 hold K={k+2,k+3}
//   B: mirrored (lanes 0-15: n=lane, K={k,k+1}; lanes 16-31: K={k+2,k+3})
//   C/D: VGPR r, lane l -> M = r + 8*(l>=16), N = l&15
// ---------------------------------------------------------------------------
__launch_bounds__(256)
__global__ void gemm_wmma_f32(const float* __restrict__ X,
                              const float* __restrict__ W,
                              const float* __restrict__ bias,
                              float* __restrict__ Y,
                              int Mrows, int accumulate) {
    __shared__ float lw[16][132];                 // padded: stride 132 dwords
    const int n0 = blockIdx.y * 16;

    for (int i = threadIdx.x; i < 16 * HDIM; i += 256) {
        const int n = i >> 7, k = i & (HDIM - 1);
        lw[n][k] = W[(size_t)(n0 + n) * HDIM + k];
    }
    __syncthreads();

    const int wave = threadIdx.x >> 5;
    const int lane = threadIdx.x & 31;
    const int m0   = blockIdx.x * 128 + wave * 16;
    if (m0 >= Mrows) return;

    const int nsub  = lane & 15;
    const int khalf = (lane >> 4) << 1;           // 0 or 2
    const float* __restrict__ xrow = X + (size_t)(m0 + nsub) * HDIM + khalf;
    const float* __restrict__ wrow = &lw[nsub][khalf];

    v8f acc = {};
#pragma unroll
    for (int k = 0; k < HDIM; k += 4) {
        v2f a, b;
        a.x = xrow[k]; a.y = xrow[k + 1];
        b.x = wrow[k]; b.y = wrow[k + 1];
        acc = __builtin_amdgcn_wmma_f32_16x16x4_f32(
            /*neg_a=*/false, a, /*neg_b=*/false, b,
            /*c_mod=*/(short)0, acc, /*reuse_a=*/false, /*reuse_b=*/false);
    }

    const float bval  = bias ? bias[n0 + nsub] : 0.0f;
    const int   mbase = m0 + ((lane >> 4) << 3);  // +8 for upper half-wave
    float* __restrict__ yp = Y + (size_t)mbase * HDIM + n0 + nsub;
#pragma unroll
    for (int r = 0; r < 8; ++r) {
        float v = acc[r] + bval;
        if (accumulate) v += yp[(size_t)r * HDIM];
        yp[(size_t)r * HDIM] = v;
    }
}

__device__ __forceinline__ float sigmoidf_(float x) {
    return 1.0f / (1.0f + __expf(-x));
}

// ---------------------------------------------------------------------------
// e[l] = v . tanh(Ah[chi[l]] + Bx[ci[l]]);  seg_max[ci[l]] = max(...)
// One wave per edge; 4 features per lane; wave32 xor-shuffle reduction.
// Segment max via monotonic-uint encoding + atomicMax.
// ---------------------------------------------------------------------------
__launch_bounds__(256)
__global__ void edge_score_kernel(const float* __restrict__ Ah,
                                  const float* __restrict__ Bx,
                                  const float* __restrict__ v_w,
                                  const int* __restrict__ ci,
                                  const int* __restrict__ chi,
                                  float* __restrict__ e,
                                  unsigned int* __restrict__ seg_max_u,
                                  int L) {
    const int edge = blockIdx.x * 8 + (threadIdx.x >> 5);
    const int lane = threadIdx.x & 31;
    if (edge >= L) return;
    const int c = ci[edge], ch = chi[edge];

    const float4 a = ((const float4*)(Ah + (size_t)ch * HDIM))[lane];
    const float4 b = ((const float4*)(Bx + (size_t)c  * HDIM))[lane];
    const float4 v = ((const float4*)v_w)[lane];

    float s = tanhf(a.x + b.x) * v.x + tanhf(a.y + b.y) * v.y
            + tanhf(a.z + b.z) * v.z + tanhf(a.w + b.w) * v.w;
#pragma unroll
    for (int off = 16; off > 0; off >>= 1) s += __shfl_xor(s, off, 32);

    if (lane == 0) {
        e[edge] = s;
        const unsigned int b32 = __float_as_uint(s);
        const unsigned int key = (b32 & 0x80000000u) ? ~b32 : (b32 | 0x80000000u);
        atomicMax(&seg_max_u[c], key);
    }
}

// x_exp[l] = exp(e[l] - seg_max[ci[l]]);  seg_sum[ci[l]] += x_exp[l]
__launch_bounds__(256)
__global__ void edge_softmax_kernel(const float* __restrict__ e,
                                    const int* __restrict__ ci,
                                    const unsigned int* __restrict__ seg_max_u,
                                    float* __restrict__ x_exp,
                                    float* __restrict__ seg_sum,
                                    int L) {
    const int t = blockIdx.x * 256 + threadIdx.x;
    if (t >= L) return;
    const int c = ci[t];
    const unsigned int key = seg_max_u[c];
    const unsigned int b32 = (key & 0x80000000u) ? (key & 0x7FFFFFFFu) : ~key;
    const float xe = __expf(e[t] - __uint_as_float(b32));
    x_exp[t] = xe;
    atomicAdd(&seg_sum[c], xe);
}

// h_hat[ci] += attn * child_h[chi];  sum_fc[ci] += sigmoid(Wfx[ci]+Ufh[chi]) * child_c[chi]
// One wave per edge; float4 per lane; f32 global atomics (L2-resident rows).
__launch_bounds__(256)
__global__ void edge_accum_kernel(const float* __restrict__ x_exp,
                                  const float* __restrict__ seg_sum,
                                  const float* __restrict__ child_h,
                                  const float* __restrict__ child_c,
                                  const float* __restrict__ Wfx,
                                  const float* __restrict__ Ufh,
                                  const int* __restrict__ ci,
                                  const int* __restrict__ chi,
                                  float* __restrict__ h_hat,
                                  float* __restrict__ sum_fc,
                                  int L) {
    const int edge = blockIdx.x * 8 + (threadIdx.x >> 5);
    const int lane = threadIdx.x & 31;
    if (edge >= L) return;
    const int c = ci[edge], ch = chi[edge];
    const float attn = x_exp[edge] / (seg_sum[c] + EPSV);

    const float4 hv = ((const float4*)(child_h + (size_t)ch * HDIM))[lane];
    const float4 cv = ((const float4*)(child_c + (size_t)ch * HDIM))[lane];
    const float4 wf = ((const float4*)(Wfx + (size_t)c  * HDIM))[lane];
    const float4 uf = ((const float4*)(Ufh + (size_t)ch * HDIM))[lane];

    float* hh = h_hat  + (size_t)c * HDIM + lane * 4;
    float* sf = sum_fc + (size_t)c * HDIM + lane * 4;

    atomicAdd(&hh[0], attn * hv.x);
    atomicAdd(&hh[1], attn * hv.y);
    atomicAdd(&hh[2], attn * hv.z);
    atomicAdd(&hh[3], attn * hv.w);

    atomicAdd(&sf[0], sigmoidf_(wf.x + uf.x) * cv.x);
    atomicAdd(&sf[1], sigmoidf_(wf.y + uf.y) * cv.y);
    atomicAdd(&sf[2], sigmoidf_(wf.z + uf.z) * cv.z);
    atomicAdd(&sf[3], sigmoidf_(wf.w + uf.w) * cv.w);
}

// Gi = Wix+Uih, Gc = Wcx+Uch, Go = Wox+Uoh already fused via accumulate-GEMM.
// c = sigmoid(Gi)*tanh(Gc) + sum_fc;  h = sigmoid(Go)*tanh(c)
__launch_bounds__(256)
__global__ void final_gate_kernel(const float* __restrict__ Gi,
                                  const float* __restrict__ Gc,
                                  const float* __restrict__ Go,
                                  const float* __restrict__ sum_fc,
                                  float* __restrict__ h_out,
                                  float* __restrict__ c_out,
                                  size_t n) {
    const size_t t = (size_t)blockIdx.x * 256 + threadIdx.x;
    if (t >= n) return;
    const float i_j = sigmoidf_(Gi[t]);
    const float c_t = tanhf(Gc[t]);
    const float c   = i_j * c_t + sum_fc[t];
    const float o_j = sigmoidf_(Go[t]);
    h_out[t] = o_j * tanhf(c);
    c_out[t] = c;
}

extern "C" void kernel_launch(void* const* d_in, const int* in_sizes, int n_in,
                              void* d_out, int out_size, void* d_ws, size_t ws_size,
                              hipStream_t stream) {
    const float* x_emb   = (const float*)d_in[0];
    const float* child_h = (const float*)d_in[1];
    const float* child_c = (const float*)d_in[2];
    const int*   ci      = (const int*)d_in[3];
    const int*   chi     = (const int*)d_in[4];
    const float* Wi_w = (const float*)d_in[5];
    const float* Ui_w = (const float*)d_in[6];
    const float* Wf_w = (const float*)d_in[7];
    const float* Uf_w = (const float*)d_in[8];
    const float* Wo_w = (const float*)d_in[9];
    const float* Uo_w = (const float*)d_in[10];
    const float* Wc_w = (const float*)d_in[11];
    const float* Uc_w = (const float*)d_in[12];
    const float* Wa_w = (const float*)d_in[13];
    const float* Ua_w = (const float*)d_in[14];
    const float* Wi_b = (const float*)d_in[15];
    const float* Wf_b = (const float*)d_in[16];
    const float* Wo_b = (const float*)d_in[17];
    const float* Wc_b = (const float*)d_in[18];
    const float* Wa_b = (const float*)d_in[19];
    const float* v_w  = (const float*)d_in[20];

    const int M = in_sizes[0] / HDIM;
    const int L = in_sizes[3];
    const size_t MH = (size_t)M * HDIM;

    // Workspace layout (9*MH + 2*L + 2*M floats ~= 230 MB)
    float* ws    = (float*)d_ws;
    float* Ah    = ws;            // child_h @ Wa^T + Wa_b
    float* Bx    = Ah  + MH;      // x_emb   @ Ua^T
    float* Gi    = Bx  + MH;      // x_emb@Wi^T + Wi_b, later += h_hat@Ui^T
    float* Wfx   = Gi  + MH;      // x_emb@Wf^T + Wf_b
    float* Go    = Wfx + MH;      // x_emb@Wo^T + Wo_b, later += h_hat@Uo^T
    float* Gc    = Go  + MH;      // x_emb@Wc^T + Wc_b, later += h_hat@Uc^T
    float* Ufh   = Gc  + MH;      // child_h @ Uf^T
    float* h_hat = Ufh + MH;
    float* sumfc = h_hat + MH;
    float* e_buf = sumfc + MH;
    float* x_exp = e_buf + L;
    float* seg_sum = x_exp + L;
    unsigned int* seg_max_u = (unsigned int*)(seg_sum + M);

    hipMemsetAsync(h_hat,     0, 2 * MH * sizeof(float), stream); // h_hat + sumfc
    hipMemsetAsync(seg_sum,   0, (size_t)M * sizeof(float), stream);
    hipMemsetAsync(seg_max_u, 0, (size_t)M * sizeof(unsigned int), stream);

    const dim3 gblk(256);
    const dim3 ggrd((M + 127) / 128, HDIM / 16);

    // Phase 1: independent projections (WMMA f32)
    gemm_wmma_f32<<<ggrd, gblk, 0, stream>>>(child_h, Wa_w, Wa_b, Ah, M, 0);
    gemm_wmma_f32<<<ggrd, gblk, 0, stream>>>(x_emb,   Ua_w, nullptr, Bx, M, 0);
    gemm_wmma_f32<<<ggrd, gblk, 0, stream>>>(x_emb,   Wi_w, Wi_b, Gi, M, 0);
    gemm_wmma_f32<<<ggrd, gblk, 0, stream>>>(x_emb,   Wf_w, Wf_b, Wfx, M, 0);
    gemm_wmma_f32<<<ggrd, gblk, 0, stream>>>(x_emb,   Wo_w, Wo_b, Go, M, 0);
    gemm_wmma_f32<<<ggrd, gblk, 0, stream>>>(x_emb,   Wc_w, Wc_b, Gc, M, 0);
    gemm_wmma_f32<<<ggrd, gblk, 0, stream>>>(child_h, Uf_w, nullptr, Ufh, M, 0);

    // Phase 2: segment-softmax attention over edges
    edge_score_kernel<<<(L + 7) / 8, gblk, 0, stream>>>(Ah, Bx, v_w, ci, chi,
                                                        e_buf, seg_max_u, L);
    edge_softmax_kernel<<<(L + 255) / 256, gblk, 0, stream>>>(e_buf, ci, seg_max_u,
                                                              x_exp, seg_sum, L);
    edge_accum_kernel<<<(L + 7) / 8, gblk, 0, stream>>>(x_exp, seg_sum,
                                                        child_h, child_c, Wfx, Ufh,
                                                        ci, chi, h_hat, sumfc, L);

    // Phase 3: U * h_hat accumulated into gate pre-activations (WMMA f32, beta=1)
    gemm_wmma_f32<<<ggrd, gblk, 0, stream>>>(h_hat, Ui_w, nullptr, Gi, M, 1);
    gemm_wmma_f32<<<ggrd, gblk, 0, stream>>>(h_hat, Uc_w, nullptr, Gc, M, 1);
    gemm_wmma_f32<<<ggrd, gblk, 0, stream>>>(h_hat, Uo_w, nullptr, Go, M, 1);

    // Phase 4: gates -> (h, c) packed into d_out
    float* h_out = (float*)d_out;
    float* c_out = h_out + MH;
    final_gate_kernel<<<(unsigned)((MH + 255) / 256), gblk, 0, stream>>>(
        Gi, Gc, Go, sumfc, h_out, c_out, MH);
}